// SpliceGraph_8933531976293
// MI455X (gfx1250) — compile-verified
//
#include <hip/hip_runtime.h>
#include <hip/hip_bf16.h>
#include <math.h>

typedef __attribute__((ext_vector_type(2))) float v2f;
typedef __attribute__((ext_vector_type(8))) float v8f;

#define N_NODES 2000
#define C_IN    4
#define L_IN    5000
#define N_EDGES 64000
#define H_DIM   512
#define REPK    207
#define KPAD    208

// ---------------------------------------------------------------------------
// Fused conv pipeline: x (N,4,5000) -> feat (N,207), one block per node.
// Stage 1: 1x1 conv 4->16 + relu + group-of-4 max  -> h1m (4 x q-tile) in LDS
// Stage 2: conv(4,4,11) stride 4 + relu            -> h2  (4 x p-tile) in LDS
// Stage 3: conv(1,4,11) stride 6 + relu            -> featP row (padded K=208)
// ---------------------------------------------------------------------------
__global__ __launch_bounds__(256) void conv_feat_kernel(
    const float* __restrict__ x,
    const float* __restrict__ w1, const float* __restrict__ b1,
    const float* __restrict__ w2, const float* __restrict__ b2,
    const float* __restrict__ w3, const float* __restrict__ b3,
    float* __restrict__ featP)
{
  __shared__ float s_h1[4][800];
  __shared__ float s_h2[4][200];
  __shared__ float s_w1[16][4];
  __shared__ float s_b1[16];
  __shared__ float s_w2[4][4][11];
  __shared__ float s_b2[4];
  __shared__ float s_w3[4][11];
  __shared__ float s_b3;

  const int tid = threadIdx.x;
  if (tid < 64)  s_w1[tid >> 2][tid & 3] = w1[tid];
  if (tid < 16)  s_b1[tid] = b1[tid];
  if (tid >= 64 && tid < 240) { int i = tid - 64; s_w2[i / 44][(i / 11) & 3][i % 11] = w2[i]; }
  if (tid >= 192 && tid < 236) { int i = tid - 192; s_w3[i / 11][i % 11] = w3[i]; }
  if (tid >= 240 && tid < 244) s_b2[tid - 240] = b2[tid - 240];
  if (tid == 255) s_b3 = b3[0];
  __syncthreads();

  const int n = blockIdx.x;
  const float* xb = x + (size_t)n * C_IN * L_IN;

  for (int t0 = 0; t0 < REPK; t0 += 32) {
    const int nt     = min(32, REPK - t0);
    const int qbase  = 24 * t0;
    const int qcount = 24 * (nt - 1) + 51;   // <= 795
    const int pcount = 6 * (nt - 1) + 11;    // <= 197

    for (int i = tid; i < qcount; i += 256) {
      const int q = qbase + i;
      const float x0 = xb[q];
      const float x1 = xb[L_IN + q];
      const float x2 = xb[2 * L_IN + q];
      const float x3 = xb[3 * L_IN + q];
      #pragma unroll
      for (int g = 0; g < 4; ++g) {
        float m = 0.f;   // relu floor: max(relu(v_j)) == max(0, max v_j)
        #pragma unroll
        for (int j = 0; j < 4; ++j) {
          const int o = g * 4 + j;
          float v = fmaf(s_w1[o][0], x0, fmaf(s_w1[o][1], x1,
                    fmaf(s_w1[o][2], x2, fmaf(s_w1[o][3], x3, s_b1[o]))));
          m = fmaxf(m, v);
        }
        s_h1[g][i] = m;
      }
    }
    __syncthreads();

    for (int i = tid; i < pcount; i += 256) {
      float a0 = s_b2[0], a1 = s_b2[1], a2 = s_b2[2], a3 = s_b2[3];
      #pragma unroll
      for (int c = 0; c < 4; ++c) {
        #pragma unroll
        for (int k = 0; k < 11; ++k) {
          const float hv = s_h1[c][4 * i + k];
          a0 = fmaf(s_w2[0][c][k], hv, a0);
          a1 = fmaf(s_w2[1][c][k], hv, a1);
          a2 = fmaf(s_w2[2][c][k], hv, a2);
          a3 = fmaf(s_w2[3][c][k], hv, a3);
        }
      }
      s_h2[0][i] = fmaxf(a0, 0.f);
      s_h2[1][i] = fmaxf(a1, 0.f);
      s_h2[2][i] = fmaxf(a2, 0.f);
      s_h2[3][i] = fmaxf(a3, 0.f);
    }
    __syncthreads();

    for (int i = tid; i < nt; i += 256) {
      float acc = s_b3;
      #pragma unroll
      for (int g = 0; g < 4; ++g)
        #pragma unroll
        for (int k = 0; k < 11; ++k)
          acc = fmaf(s_w3[g][k], s_h2[g][6 * i + k], acc);
      featP[(size_t)n * KPAD + t0 + i] = fmaxf(acc, 0.f);
    }
    __syncthreads();
  }
  if (tid == 0) featP[(size_t)n * KPAD + REPK] = 0.f;   // K pad column
}

// ---------------------------------------------------------------------------
// WMMA f32 GEMM: C(M x N) = A(M x K) * B(K x N), all row-major.
// One wave per 16x64 output strip: 4 accumulators share each A fragment
// (4x arithmetic intensity on A traffic).  K unrolled by 2 (K % 8 == 0 for
// all call sites: 208, 208, 512).  V_WMMA_F32_16X16X4_F32.
// A frag: lane%16 = M row, lane/16 -> K pair {0,1}/{2,3} (v0=k, v1=k+1)
// B frag mirrors A (lane%16 = N col).  C/D: row = i + 8*(lane>=16).
// ---------------------------------------------------------------------------
__global__ __launch_bounds__(256) void gemm_wmma_kernel(
    const float* __restrict__ A, int lda,
    const float* __restrict__ B, int ldb,
    float* __restrict__ Cmat, int ldc,
    int Mtiles, int NtilesG,   // NtilesG = (N/16)/4 groups of 4 N-tiles
    int K)
{
  const int wave = (int)((blockIdx.x * blockDim.x + threadIdx.x) >> 5);
  const int lane = threadIdx.x & 31;
  if (wave >= Mtiles * NtilesG) return;
  const int tm = wave / NtilesG;
  const int tg = wave - tm * NtilesG;
  const int mrow  = tm * 16 + (lane & 15);
  const int ncol  = tg * 64 + (lane & 15);
  const int khalf = (lane >> 4) << 1;          // 0 or 2

  v8f acc0 = {0.f,0.f,0.f,0.f,0.f,0.f,0.f,0.f};
  v8f acc1 = acc0, acc2 = acc0, acc3 = acc0;

  const float* arow = A + (size_t)mrow * lda + khalf;
  const float* bcol = B + (size_t)khalf * ldb + ncol;

  for (int k0 = 0; k0 < K; k0 += 8) {
    __builtin_prefetch(arow + k0 + 64, 0, 1);   // global_prefetch_b8
    #pragma unroll
    for (int u = 0; u < 8; u += 4) {
      v2f a;
      a.x = arow[k0 + u];
      a.y = arow[k0 + u + 1];
      const float* brow0 = bcol + (size_t)(k0 + u) * ldb;
      const float* brow1 = brow0 + ldb;
      v2f b0; b0.x = brow0[0];  b0.y = brow1[0];
      v2f b1; b1.x = brow0[16]; b1.y = brow1[16];
      v2f b2; b2.x = brow0[32]; b2.y = brow1[32];
      v2f b3; b3.x = brow0[48]; b3.y = brow1[48];
      acc0 = __builtin_amdgcn_wmma_f32_16x16x4_f32(false, a, false, b0, (short)0, acc0, false, false);
      acc1 = __builtin_amdgcn_wmma_f32_16x16x4_f32(false, a, false, b1, (short)0, acc1, false, false);
      acc2 = __builtin_amdgcn_wmma_f32_16x16x4_f32(false, a, false, b2, (short)0, acc2, false, false);
      acc3 = __builtin_amdgcn_wmma_f32_16x16x4_f32(false, a, false, b3, (short)0, acc3, false, false);
    }
  }

  const int row0 = tm * 16 + ((lane >> 4) << 3);
  #pragma unroll
  for (int i = 0; i < 8; ++i) {
    float* crow = Cmat + (size_t)(row0 + i) * ldc + ncol;
    crow[0]  = acc0[i];
    crow[16] = acc1[i];
    crow[32] = acc2[i];
    crow[48] = acc3[i];
  }
}

// ---------------------------------------------------------------------------
// Weight padding:  (207,512) -> (208,512) with zero last row, both matrices.
// ---------------------------------------------------------------------------
__global__ void pad_w_kernel(const float* __restrict__ gw, const float* __restrict__ lw,
                             float* __restrict__ gwp, float* __restrict__ lwp)
{
  const int idx = blockIdx.x * blockDim.x + threadIdx.x;
  if (idx < KPAD * H_DIM) {
    const int k = idx / H_DIM;
    gwp[idx] = (k < REPK) ? gw[idx] : 0.f;
    lwp[idx] = (k < REPK) ? lw[idx] : 0.f;
  }
}

// ---------------------------------------------------------------------------
// Graph aggregation helpers
// ---------------------------------------------------------------------------
__global__ void zero_f32_kernel(float* p, int n) {
  const int i = blockIdx.x * blockDim.x + threadIdx.x;
  if (i < n) p[i] = 0.f;
}

__global__ void deg_count_kernel(const int* __restrict__ ei, float* __restrict__ deg) {
  const int e = blockIdx.x * blockDim.x + threadIdx.x;
  if (e < N_EDGES) atomicAdd(&deg[ei[N_EDGES + e]], 1.0f);
}

__global__ void dinv_kernel(float* deg) {
  const int i = blockIdx.x * blockDim.x + threadIdx.x;
  if (i < N_NODES) deg[i] = rsqrtf(deg[i] + 1.0f);   // +1 for self loop
}

// agg initialized with the self-loop contribution h[n]*dinv[n]^2
__global__ void selfloop_kernel(const float* __restrict__ h, const float* __restrict__ dinv,
                                float* __restrict__ agg) {
  const int idx = blockIdx.x * blockDim.x + threadIdx.x;
  if (idx < N_NODES * H_DIM) {
    const float di = dinv[idx >> 9];
    agg[idx] = h[idx] * di * di;
  }
}

__global__ void edge_agg_kernel(const int* __restrict__ ei, const float* __restrict__ h,
                                const float* __restrict__ dinv, float* __restrict__ agg) {
  const long long idx = (long long)blockIdx.x * blockDim.x + threadIdx.x;
  if (idx < (long long)N_EDGES * H_DIM) {
    const int e = (int)(idx >> 9);
    const int f = (int)(idx & 511);
    const int s = ei[e];
    const int d = ei[N_EDGES + e];
    const float norm = dinv[s] * dinv[d];
    atomicAdd(&agg[(size_t)d * H_DIM + f], h[(size_t)s * H_DIM + f] * norm);
  }
}

__global__ void tanh_bias_kernel(float* z, const float* __restrict__ b) {
  const int idx = blockIdx.x * blockDim.x + threadIdx.x;
  if (idx < N_NODES * H_DIM) z[idx] = tanhf(z[idx] + b[idx & 511]);
}

// out = relu((1-g)*(xl+lin_b) + g*z), g = sigmoid(gpre + gate_b)
__global__ void combine_kernel(const float* __restrict__ z, const float* __restrict__ gpre,
                               const float* __restrict__ gate_b,
                               const float* __restrict__ xl, const float* __restrict__ lin_b,
                               float* __restrict__ out) {
  const int idx = blockIdx.x * blockDim.x + threadIdx.x;
  if (idx < N_NODES * H_DIM) {
    const int f = idx & 511;
    const float zv = z[idx];
    const float gv = 1.f / (1.f + expf(-(gpre[idx] + gate_b[f])));
    const float xv = xl[idx] + lin_b[f];
    out[idx] = fmaxf((1.f - gv) * xv + gv * zv, 0.f);
  }
}

// ---------------------------------------------------------------------------
// BatchNorm over axis 0: one block per feature, in-place on d_out.
// ---------------------------------------------------------------------------
__global__ __launch_bounds__(256) void bn_kernel(float* __restrict__ out,
    const float* __restrict__ gamma, const float* __restrict__ beta)
{
  const int f = blockIdx.x;
  const int tid = threadIdx.x;
  __shared__ float ss[256], sq[256];
  float s = 0.f, q = 0.f;
  for (int n = tid; n < N_NODES; n += 256) {
    const float v = out[(size_t)n * H_DIM + f];
    s += v; q += v * v;
  }
  ss[tid] = s; sq[tid] = q;
  __syncthreads();
  for (int off = 128; off > 0; off >>= 1) {
    if (tid < off) { ss[tid] += ss[tid + off]; sq[tid] += sq[tid + off]; }
    __syncthreads();
  }
  const float mean  = ss[0] * (1.0f / N_NODES);
  const float var   = sq[0] * (1.0f / N_NODES) - mean * mean;
  const float scale = rsqrtf(var + 1e-5f) * gamma[f];
  const float shift = beta[f] - mean * scale;
  for (int n = tid; n < N_NODES; n += 256) {
    const size_t i = (size_t)n * H_DIM + f;
    out[i] = out[i] * scale + shift;
  }
}

// ---------------------------------------------------------------------------
extern "C" void kernel_launch(void* const* d_in, const int* in_sizes, int n_in,
                              void* d_out, int out_size, void* d_ws, size_t ws_size,
                              hipStream_t stream) {
  const float* x      = (const float*)d_in[0];
  const int*   ei     = (const int*)  d_in[1];
  const float* w1     = (const float*)d_in[2];
  const float* b1     = (const float*)d_in[3];
  const float* w2     = (const float*)d_in[4];
  const float* b2     = (const float*)d_in[5];
  const float* w3     = (const float*)d_in[6];
  const float* b3     = (const float*)d_in[7];
  const float* gcn_w  = (const float*)d_in[8];
  const float* gcn_b  = (const float*)d_in[9];
  const float* lin_w  = (const float*)d_in[10];
  const float* lin_b  = (const float*)d_in[11];
  const float* gate_w = (const float*)d_in[12];
  const float* gate_b = (const float*)d_in[13];
  const float* bn_g   = (const float*)d_in[14];
  const float* bn_b   = (const float*)d_in[15];
  float* out = (float*)d_out;

  // workspace layout (floats), total ~18.9 MB
  float* ws    = (float*)d_ws;
  float* featP = ws;                          // 2000*208
  float* gwp   = featP + N_NODES * KPAD;      // 208*512
  float* lwp   = gwp   + KPAD * H_DIM;        // 208*512
  float* hbuf  = lwp   + KPAD * H_DIM;        // 2000*512
  float* xlbuf = hbuf  + N_NODES * H_DIM;     // 2000*512
  float* zbuf  = xlbuf + N_NODES * H_DIM;     // 2000*512
  float* gbuf  = zbuf  + N_NODES * H_DIM;     // 2000*512
  float* deg   = gbuf  + N_NODES * H_DIM;     // 2000

  const int NH = N_NODES * H_DIM;

  pad_w_kernel<<<(KPAD * H_DIM + 255) / 256, 256, 0, stream>>>(gcn_w, lin_w, gwp, lwp);
  conv_feat_kernel<<<N_NODES, 256, 0, stream>>>(x, w1, b1, w2, b2, w3, b3, featP);

  const int Mt  = N_NODES / 16;        // 125 M-tiles
  const int NtG = (H_DIM / 16) / 4;    // 8 groups of 4 N-tiles
  const int gemm_blocks = (Mt * NtG * 32 + 255) / 256;
  gemm_wmma_kernel<<<gemm_blocks, 256, 0, stream>>>(featP, KPAD, gwp, H_DIM, hbuf,  H_DIM, Mt, NtG, KPAD);
  gemm_wmma_kernel<<<gemm_blocks, 256, 0, stream>>>(featP, KPAD, lwp, H_DIM, xlbuf, H_DIM, Mt, NtG, KPAD);

  zero_f32_kernel<<<(N_NODES + 255) / 256, 256, 0, stream>>>(deg, N_NODES);
  deg_count_kernel<<<(N_EDGES + 255) / 256, 256, 0, stream>>>(ei, deg);
  dinv_kernel<<<(N_NODES + 255) / 256, 256, 0, stream>>>(deg);
  selfloop_kernel<<<(NH + 255) / 256, 256, 0, stream>>>(hbuf, deg, zbuf);
  const long long ef = (long long)N_EDGES * H_DIM;
  edge_agg_kernel<<<(unsigned)((ef + 255) / 256), 256, 0, stream>>>(ei, hbuf, deg, zbuf);
  tanh_bias_kernel<<<(NH + 255) / 256, 256, 0, stream>>>(zbuf, gcn_b);

  gemm_wmma_kernel<<<gemm_blocks, 256, 0, stream>>>(zbuf, H_DIM, gate_w, H_DIM, gbuf, H_DIM, Mt, NtG, H_DIM);

  combine_kernel<<<(NH + 255) / 256, 256, 0, stream>>>(zbuf, gbuf, gate_b, xlbuf, lin_b, out);
  bn_kernel<<<H_DIM, 256, 0, stream>>>(out, bn_g, bn_b);
}